// SE_44547400794189
// MI455X (gfx1250) — compile-verified
//
#include <hip/hip_runtime.h>
#include <hip/hip_bf16.h>

typedef __attribute__((ext_vector_type(2))) float v2f;
typedef __attribute__((ext_vector_type(4))) float v4f;
typedef __attribute__((ext_vector_type(8))) float v8f;

#define HW_ELEMS   16384      // 128*128
#define HW_VEC4    4096       // HW_ELEMS / 4
#define NCH        256        // channels
#define NB         16         // batch

// ---------------------------------------------------------------------------
// Kernel 1: global average pool.  One 256-thread block per (b,c) plane.
// ---------------------------------------------------------------------------
__global__ __launch_bounds__(256) void se_pool_kernel(
    const float* __restrict__ x, float* __restrict__ wbuf)
{
    const int plane = blockIdx.x;                       // 0 .. 4095
    const v4f* __restrict__ xp =
        (const v4f*)(x + (size_t)plane * HW_ELEMS);

    float s = 0.0f;
    #pragma unroll 4
    for (int i = threadIdx.x; i < HW_VEC4; i += 256) {
        v4f v = xp[i];
        s += (v.x + v.y) + (v.z + v.w);
    }

    // wave32 reduction
    for (int off = 16; off > 0; off >>= 1)
        s += __shfl_down(s, off, 32);

    __shared__ float red[8];
    const int lane = threadIdx.x & 31;
    const int wv   = threadIdx.x >> 5;
    if (lane == 0) red[wv] = s;
    __syncthreads();

    if (threadIdx.x < 8) {
        float t = red[threadIdx.x];
        for (int off = 4; off > 0; off >>= 1)
            t += __shfl_down(t, off, 8);
        if (threadIdx.x == 0)
            wbuf[plane] = t * (1.0f / (float)HW_ELEMS);
    }
}

// ---------------------------------------------------------------------------
// Kernel 2: tiny MLP with fp32 WMMA (V_WMMA_F32_16X16X4_F32).
// One block of 512 threads = 16 waves; wave w owns output columns
// [16*w, 16*w+16).  h = swish(w @ W1^T + b1); g = sigmoid(h @ W2^T + b2).
//
// Per-lane fragment layout (ISA 7.12.2, wave32):
//   A (16x4 f32):  row = lane&15,  a.x = A[row][k0 + 2*hi], a.y = +1 (hi = lane>=16)
//   B (4x16 f32):  col = lane&15,  b.x = B[k0 + 2*hi][col], b.y = B[k0+1+2*hi][col]
//                  with B[k][n] = W[n][k]  ->  contiguous float2 of row-major W.
//   C/D (16x16):   acc[r] -> row r + 8*hi, col lane&15.
// ---------------------------------------------------------------------------
__global__ __launch_bounds__(512) void se_mlp_kernel(
    const float* __restrict__ wbuf,   // [16,256] pooled means
    const float* __restrict__ W1,     // [256,256] row-major
    const float* __restrict__ b1,     // [256]
    const float* __restrict__ W2,     // [256,256] row-major
    const float* __restrict__ b2,     // [256]
    float* __restrict__ gbuf)         // [16,256] gate out
{
    __shared__ float h_lds[NB * NCH];               // 16 KB

    const int lane  = threadIdx.x & 31;
    const int wave  = threadIdx.x >> 5;             // 0..15 -> N tile
    const int hi    = lane >> 4;                    // 0 or 1
    const int row   = lane & 15;                    // A-fragment row / D column idx
    const int n     = wave * 16 + row;              // absolute output column

    // ---- GEMM 1: h = wbuf @ W1^T ----
    v8f acc = {};
    #pragma unroll 4
    for (int k0 = 0; k0 < NCH; k0 += 4) {
        const int ka = k0 + hi * 2;
        v2f a = *(const v2f*)(wbuf + row * NCH + ka);
        v2f b = *(const v2f*)(W1   + n   * NCH + ka);
        acc = __builtin_amdgcn_wmma_f32_16x16x4_f32(
                  false, a, false, b, (short)0, acc, false, false);
    }

    const float bias1 = b1[n];
    #pragma unroll
    for (int r = 0; r < 8; ++r) {
        const int m = r + hi * 8;
        float hv = acc[r] + bias1;
        hv = hv / (1.0f + expf(-hv));               // swish
        h_lds[m * NCH + n] = hv;
    }
    __syncthreads();

    // ---- GEMM 2: g = sigmoid(h @ W2^T + b2) ----
    v8f acc2 = {};
    #pragma unroll 4
    for (int k0 = 0; k0 < NCH; k0 += 4) {
        const int ka = k0 + hi * 2;
        v2f a = *(const v2f*)(&h_lds[row * NCH + ka]);
        v2f b = *(const v2f*)(W2 + n * NCH + ka);
        acc2 = __builtin_amdgcn_wmma_f32_16x16x4_f32(
                   false, a, false, b, (short)0, acc2, false, false);
    }

    const float bias2 = b2[n];
    #pragma unroll
    for (int r = 0; r < 8; ++r) {
        const int m = r + hi * 8;
        float gv = acc2[r] + bias2;
        gv = 1.0f / (1.0f + expf(-gv));             // sigmoid
        gbuf[m * NCH + n] = gv;
    }
}

// ---------------------------------------------------------------------------
// Kernel 3: out = g[b,c] * x.  Pure bandwidth: float4 per thread,
// non-temporal store (output is write-once) so x's L2 residency survives.
// ---------------------------------------------------------------------------
__global__ __launch_bounds__(256) void se_scale_kernel(
    const float* __restrict__ x,
    const float* __restrict__ gbuf,
    float* __restrict__ out)
{
    const size_t i = (size_t)blockIdx.x * 256 + threadIdx.x;  // v4f index
    const float gv = gbuf[i >> 12];                           // 4096 v4f / plane
    v4f v = ((const v4f*)x)[i];
    v *= gv;
    __builtin_nontemporal_store(v, (v4f*)out + i);
}

// ---------------------------------------------------------------------------
// Launch: pool -> MLP -> scale (stream-ordered).
// ---------------------------------------------------------------------------
extern "C" void kernel_launch(void* const* d_in, const int* in_sizes, int n_in,
                              void* d_out, int out_size, void* d_ws, size_t ws_size,
                              hipStream_t stream)
{
    const float* x  = (const float*)d_in[0];
    const float* W1 = (const float*)d_in[1];
    const float* b1 = (const float*)d_in[2];
    const float* W2 = (const float*)d_in[3];
    const float* b2 = (const float*)d_in[4];
    float* out = (float*)d_out;

    float* wbuf = (float*)d_ws;          // 16*256 floats
    float* gbuf = wbuf + NB * NCH;       // 16*256 floats

    se_pool_kernel <<<NB * NCH, 256, 0, stream>>>(x, wbuf);
    se_mlp_kernel  <<<1, 512, 0, stream>>>(wbuf, W1, b1, W2, b2, gbuf);

    const int total_vec4 = NB * NCH * HW_VEC4;     // 16,777,216 v4fs
    se_scale_kernel<<<total_vec4 / 256, 256, 0, stream>>>(x, gbuf, out);
}